// MultiBoxLoss_34394098106624
// MI455X (gfx1250) — compile-verified
//
#include <hip/hip_runtime.h>
#include <hip/hip_bf16.h>

#define BB   16
#define PP   131072
#define NP   (BB*PP)        // 2097152 anchors
#define BLK  256
#define APT  16             // anchors per thread
#define APB  (BLK*APT)      // 4096 anchors per block
#define NBLK (NP/APB)       // 512 blocks, 32 per batch row
#define BPR  (PP/APB)       // 32 blocks per row
#define NWAVE (BLK/32)      // 8 waves per block

typedef __attribute__((ext_vector_type(8)))  float    v8f;
typedef __attribute__((ext_vector_type(2)))  float    v2f;

#if __has_builtin(__builtin_amdgcn_wmma_f32_16x16x4_f32)
#define HAVE_WMMA_F32X4 1
#endif

// ---------- wave-level exact fp32 sum via WMMA (codegen-confirmed on gfx1250) ----------
// A (16x4 f32): lane L<16 -> row M=L holds (K0,K1); lane L>=16 -> row M=L-16 holds (K2,K3).
// With a=(v,0) and B=ones: D[m][n] = v_m + v_{m+16}. Per-lane sum of the 8 D VGPRs
// covers rows {0..7} (lanes 0-15) or {8..15} (lanes 16-31); xor-16 add -> full wave sum.
// Exact fp32 FMA semantics -> also exact for integer-valued floats < 2^24.
__device__ __forceinline__ float wave_sum_f32(float v) {
#ifdef HAVE_WMMA_F32X4
  v2f a;  a[0] = v;    a[1] = 0.0f;
  v2f bm; bm[0] = 1.0f; bm[1] = 1.0f;
  v8f c;
#pragma unroll
  for (int i = 0; i < 8; ++i) c[i] = 0.0f;
  c = __builtin_amdgcn_wmma_f32_16x16x4_f32(false, a, false, bm, (short)0, c, false, false);
  float h = ((c[0] + c[1]) + (c[2] + c[3])) + ((c[4] + c[5]) + (c[6] + c[7]));
  h += __shfl_xor(h, 16, 32);
  return h;
#else
#pragma unroll
  for (int m = 16; m >= 1; m >>= 1) v += __shfl_xor(v, m, 32);
  return v;
#endif
}

__device__ __forceinline__ float sl1(float x) {
  float a = fabsf(x);
  return a < 1.0f ? 0.5f * x * x : a - 0.5f;
}

// ce_neg (positives zeroed) and its orderable bit pattern (ce >= 0 always).
__device__ __forceinline__ unsigned cen_bits_of(int label, float2 c, float* cen_out) {
  float mx  = fmaxf(c.x, c.y);
  float d   = fabsf(c.x - c.y);
  float lse = mx + log1pf(expf(-d));
  float cen = (label != 0) ? 0.0f : (lse - c.x);
  *cen_out = cen;
  return __float_as_uint(cen);
}

// ---------------- K1: main streaming pass ----------------
__global__ __launch_bounds__(BLK) void k1_main(
    const float* __restrict__ loc_data, const float* __restrict__ conf_data,
    const float* __restrict__ landm_data, const float* __restrict__ loc_t,
    const float* __restrict__ landm_t, const int* __restrict__ conf_t,
    unsigned* __restrict__ hist1, unsigned* __restrict__ num_pos,
    unsigned* __restrict__ n1cnt,
    float* __restrict__ part_l, float* __restrict__ part_lm,
    float* __restrict__ part_pce)
{
  // Per-wave privatized histograms: kills cross-wave same-address DS-atomic
  // serialization on the hot, concentrated ce bins. 8*1024*4B = 32 KB of 320 KB LDS.
  __shared__ unsigned lh[NWAVE][1024];
  __shared__ float wl_[NWAVE], wlm_[NWAVE], wpce_[NWAVE], wnp_[NWAVE], wn1_[NWAVE];
  const int tid  = threadIdx.x;
  const int blk  = blockIdx.x;
  const int b    = blk / BPR;
  const int base = blk * APB;
  const int wave = tid >> 5, lane = tid & 31;
  for (int i = tid; i < NWAVE * 1024; i += BLK) ((unsigned*)lh)[i] = 0u;
  __syncthreads();

  float acc_l = 0.f, acc_lm = 0.f, acc_pce = 0.f;
  unsigned np = 0u, n1 = 0u;
  const float2* cd = (const float2*)conf_data;
  unsigned* mh = lh[wave];

  for (int j = 0; j < APT; ++j) {
    int idx = base + j * BLK + tid;
    if (j + 1 < APT) {
      __builtin_prefetch(&cd[idx + BLK], 0, 1);
      __builtin_prefetch(&conf_t[idx + BLK], 0, 1);
    }
    int label = conf_t[idx];
    float2 c  = cd[idx];
    float mx  = fmaxf(c.x, c.y);
    float d   = fabsf(c.x - c.y);
    float lse = mx + log1pf(expf(-d));
    bool pos  = (label != 0);
    float picked = pos ? c.y : c.x;
    float ce  = lse - picked;
    float cen = pos ? 0.0f : ce;
    unsigned bits = __float_as_uint(cen);
    atomicAdd(&mh[bits >> 22], 1u);
    if (pos) {
      np++; acc_pce += ce;
      float4 a4 = ((const float4*)loc_data)[idx];
      float4 t4 = ((const float4*)loc_t)[idx];
      acc_l += sl1(a4.x - t4.x) + sl1(a4.y - t4.y) + sl1(a4.z - t4.z) + sl1(a4.w - t4.w);
    }
    if (label > 0) {
      n1++;
      const float2* pd = (const float2*)(landm_data + (size_t)idx * 10);
      const float2* pt = (const float2*)(landm_t   + (size_t)idx * 10);
#pragma unroll
      for (int q = 0; q < 5; ++q) {
        float2 a2 = pd[q], t2 = pt[q];
        acc_lm += sl1(a2.x - t2.x) + sl1(a2.y - t2.y);
      }
    }
  }

  // Uniform control flow here: EXEC all ones for WMMA reductions.
  float rl   = wave_sum_f32(acc_l);
  float rlm  = wave_sum_f32(acc_lm);
  float rpce = wave_sum_f32(acc_pce);
  float rnp  = wave_sum_f32((float)np);   // exact: integer sums << 2^24
  float rn1  = wave_sum_f32((float)n1);
  if (lane == 0) { wl_[wave] = rl; wlm_[wave] = rlm; wpce_[wave] = rpce;
                   wnp_[wave] = rnp; wn1_[wave] = rn1; }
  __syncthreads();
  // Merge 8 private histograms -> global (one atomic per nonzero bin).
  for (int i = tid; i < 1024; i += BLK) {
    unsigned v = 0;
#pragma unroll
    for (int w = 0; w < NWAVE; ++w) v += lh[w][i];
    if (v) atomicAdd(&hist1[b * 1024 + i], v);
  }
  if (tid == 0) {
    float sl = 0, slm = 0, spce = 0, snp = 0, sn1 = 0;
#pragma unroll
    for (int w = 0; w < NWAVE; ++w) { sl += wl_[w]; slm += wlm_[w]; spce += wpce_[w];
                                      snp += wnp_[w]; sn1 += wn1_[w]; }
    part_l[blk] = sl; part_lm[blk] = slm; part_pce[blk] = spce;
    atomicAdd(&num_pos[b], (unsigned)(snp + 0.5f));
    atomicAdd(n1cnt, (unsigned)(sn1 + 0.5f));
  }
}

// ---------------- radix-select scan kernels (tiny) ----------------
__global__ void k_scan1(const unsigned* __restrict__ hist1,
                        const unsigned* __restrict__ num_pos,
                        unsigned* __restrict__ bin1, unsigned* __restrict__ rem1,
                        unsigned* __restrict__ kzero)
{
  int b = threadIdx.x; if (b >= BB) return;
  long long k = 7LL * (long long)num_pos[b];
  if (k > PP - 1) k = PP - 1;
  if (k <= 0) { kzero[b] = 1u; bin1[b] = 0u; rem1[b] = 0u; return; }
  kzero[b] = 0u;
  const unsigned* h = hist1 + b * 1024;
  unsigned long long cum = 0;
  for (int g = 1023; g >= 0; --g) {
    unsigned long long c2 = cum + h[g];
    if (c2 >= (unsigned long long)k) { bin1[b] = (unsigned)g; rem1[b] = (unsigned)(k - cum); return; }
    cum = c2;
  }
}

__global__ void k_scan2(const unsigned* __restrict__ hist2,
                        const unsigned* __restrict__ rem1,
                        const unsigned* __restrict__ kzero,
                        unsigned* __restrict__ bin2, unsigned* __restrict__ rem2)
{
  int b = threadIdx.x; if (b >= BB) return;
  if (kzero[b]) { bin2[b] = 0u; rem2[b] = 0u; return; }
  const unsigned* h = hist2 + b * 2048;
  unsigned k = rem1[b], cum = 0;
  for (int g = 2047; g >= 0; --g) {
    unsigned c2 = cum + h[g];
    if (c2 >= k) { bin2[b] = (unsigned)g; rem2[b] = k - cum; return; }
    cum = c2;
  }
}

__global__ void k_scan3(const unsigned* __restrict__ hist3,
                        const unsigned* __restrict__ bin1, const unsigned* __restrict__ bin2,
                        const unsigned* __restrict__ rem2, const unsigned* __restrict__ kzero,
                        unsigned* __restrict__ tbits, unsigned* __restrict__ rem3)
{
  int b = threadIdx.x; if (b >= BB) return;
  if (kzero[b]) { tbits[b] = 0xFFFFFFFFu; rem3[b] = 0u; return; }
  const unsigned* h = hist3 + b * 2048;
  unsigned k = rem2[b], cum = 0;
  for (int g = 2047; g >= 0; --g) {
    unsigned c2 = cum + h[g];
    if (c2 >= k) {
      tbits[b] = (bin1[b] << 22) | (bin2[b] << 11) | (unsigned)g;
      rem3[b]  = k - cum;   // elements equal to exact threshold to include
      return;
    }
    cum = c2;
  }
}

// ---------------- refinement histogram passes (L2-resident conf) ----------------
__global__ __launch_bounds__(BLK) void k_hist2(
    const float* __restrict__ conf_data, const int* __restrict__ conf_t,
    const unsigned* __restrict__ bin1, const unsigned* __restrict__ kzero,
    unsigned* __restrict__ hist2)
{
  int blk = blockIdx.x, b = blk / BPR;
  if (kzero[b]) return;
  unsigned g1 = bin1[b];
  int base = blk * APB;
  const float2* cd = (const float2*)conf_data;
  for (int j = 0; j < APT; ++j) {
    int idx = base + j * BLK + threadIdx.x;
    float cen; unsigned bits = cen_bits_of(conf_t[idx], cd[idx], &cen);
    if ((bits >> 22) == g1) atomicAdd(&hist2[b * 2048 + ((bits >> 11) & 0x7FFu)], 1u);
  }
}

__global__ __launch_bounds__(BLK) void k_hist3(
    const float* __restrict__ conf_data, const int* __restrict__ conf_t,
    const unsigned* __restrict__ bin1, const unsigned* __restrict__ bin2,
    const unsigned* __restrict__ kzero, unsigned* __restrict__ hist3)
{
  int blk = blockIdx.x, b = blk / BPR;
  if (kzero[b]) return;
  unsigned pre = (bin1[b] << 11) | bin2[b];
  int base = blk * APB;
  const float2* cd = (const float2*)conf_data;
  for (int j = 0; j < APT; ++j) {
    int idx = base + j * BLK + threadIdx.x;
    float cen; unsigned bits = cen_bits_of(conf_t[idx], cd[idx], &cen);
    if ((bits >> 11) == pre) atomicAdd(&hist3[b * 2048 + (bits & 0x7FFu)], 1u);
  }
}

// ---------------- K7: sum of values strictly above threshold ----------------
__global__ __launch_bounds__(BLK) void k_sumgt(
    const float* __restrict__ conf_data, const int* __restrict__ conf_t,
    const unsigned* __restrict__ tbits, const unsigned* __restrict__ kzero,
    float* __restrict__ part_gt)
{
  __shared__ float wsum[NWAVE];
  int blk = blockIdx.x, b = blk / BPR, tid = threadIdx.x;
  float acc = 0.f;
  if (!kzero[b]) {                     // block-uniform branch
    unsigned t = tbits[b];
    int base = blk * APB;
    const float2* cd = (const float2*)conf_data;
    for (int j = 0; j < APT; ++j) {
      int idx = base + j * BLK + tid;
      if (j + 1 < APT) __builtin_prefetch(&cd[idx + BLK], 0, 1);
      float cen; unsigned bits = cen_bits_of(conf_t[idx], cd[idx], &cen);
      if (bits > t) acc += cen;
    }
  }
  float r = wave_sum_f32(acc);         // EXEC full here
  if ((tid & 31) == 0) wsum[tid >> 5] = r;
  __syncthreads();
  if (tid == 0) {
    float s2 = 0;
#pragma unroll
    for (int w = 0; w < NWAVE; ++w) s2 += wsum[w];
    part_gt[blk] = s2;
  }
}

// ---------------- K8: finalize (fixed-order, deterministic) ----------------
__global__ __launch_bounds__(512) void k_final(
    const float* __restrict__ part_l, const float* __restrict__ part_lm,
    const float* __restrict__ part_pce, const float* __restrict__ part_gt,
    const unsigned* __restrict__ num_pos, const unsigned* __restrict__ n1cnt,
    const unsigned* __restrict__ tbits, const unsigned* __restrict__ rem3,
    const unsigned* __restrict__ kzero, float* __restrict__ out)
{
  __shared__ float sl[512], slm[512], spce[512], sgt[512];
  int t = threadIdx.x;
  sl[t] = part_l[t]; slm[t] = part_lm[t]; spce[t] = part_pce[t]; sgt[t] = part_gt[t];
  __syncthreads();
  for (int s = 256; s > 0; s >>= 1) {
    if (t < s) { sl[t] += sl[t + s]; slm[t] += slm[t + s];
                 spce[t] += spce[t + s]; sgt[t] += sgt[t + s]; }
    __syncthreads();
  }
  if (t == 0) {
    unsigned npos_tot = 0;
    for (int b = 0; b < BB; ++b) npos_tot += num_pos[b];
    float tie = 0.f;
    for (int b = 0; b < BB; ++b)
      if (!kzero[b]) tie += (float)rem3[b] * __uint_as_float(tbits[b]);
    float N  = fmaxf((float)npos_tot, 1.0f);
    float N1 = fmaxf((float)(*n1cnt), 1.0f);
    out[0] = sl[0] / N;                       // loss_l / N
    out[1] = (spce[0] + sgt[0] + tie) / N;    // loss_c / N
    out[2] = slm[0] / N1;                     // loss_landm / N1
  }
}

extern "C" void kernel_launch(void* const* d_in, const int* in_sizes, int n_in,
                              void* d_out, int out_size, void* d_ws, size_t ws_size,
                              hipStream_t stream) {
  (void)in_sizes; (void)n_in; (void)out_size; (void)ws_size;
  const float* loc_data   = (const float*)d_in[0];
  const float* conf_data  = (const float*)d_in[1];
  const float* landm_data = (const float*)d_in[2];
  const float* loc_t      = (const float*)d_in[3];
  const float* landm_t    = (const float*)d_in[4];
  const int*   conf_t     = (const int*)d_in[5];

  unsigned* u       = (unsigned*)d_ws;
  unsigned* hist1   = u;                 // 16*1024
  unsigned* hist2   = u + 16384;         // 16*2048
  unsigned* hist3   = u + 49152;         // 16*2048
  unsigned* num_pos = u + 81920;         // 16
  unsigned* n1cnt   = u + 81936;         // 1 (padded)
  unsigned* bin1    = u + 81952;
  unsigned* rem1    = u + 81968;
  unsigned* bin2    = u + 81984;
  unsigned* rem2    = u + 82000;
  unsigned* tbits   = u + 82016;
  unsigned* rem3    = u + 82032;
  unsigned* kzero   = u + 82048;         // 16 -> ends 82064
  float* part_l   = (float*)(u + 82064);
  float* part_lm  = part_l  + NBLK;
  float* part_pce = part_lm + NBLK;
  float* part_gt  = part_pce + NBLK;
  size_t ws_used = (size_t)(82064 + 4 * NBLK) * sizeof(unsigned);

  hipMemsetAsync(d_ws, 0, ws_used, stream);
  k1_main<<<NBLK, BLK, 0, stream>>>(loc_data, conf_data, landm_data, loc_t, landm_t,
                                    conf_t, hist1, num_pos, n1cnt,
                                    part_l, part_lm, part_pce);
  k_scan1<<<1, 16, 0, stream>>>(hist1, num_pos, bin1, rem1, kzero);
  k_hist2<<<NBLK, BLK, 0, stream>>>(conf_data, conf_t, bin1, kzero, hist2);
  k_scan2<<<1, 16, 0, stream>>>(hist2, rem1, kzero, bin2, rem2);
  k_hist3<<<NBLK, BLK, 0, stream>>>(conf_data, conf_t, bin1, bin2, kzero, hist3);
  k_scan3<<<1, 16, 0, stream>>>(hist3, bin1, bin2, rem2, kzero, tbits, rem3);
  k_sumgt<<<NBLK, BLK, 0, stream>>>(conf_data, conf_t, tbits, kzero, part_gt);
  k_final<<<1, 512, 0, stream>>>(part_l, part_lm, part_pce, part_gt,
                                 num_pos, n1cnt, tbits, rem3, kzero, (float*)d_out);
}